// ProteinFeatures_23476291239999
// MI455X (gfx1250) — compile-verified
//
#include <hip/hip_runtime.h>
#include <hip/hip_bf16.h>
#include <math.h>

// ---------------------------------------------------------------------------
// ProteinFeatures for MI455X (gfx1250, wave32).
//   B=2, N=4096, TOP_K=30, NUM_RBF=16, NUM_PE=16, hidden=128
// Linear layers run on V_WMMA_F32_16X16X4_F32 (full fp32 matrix pipe),
// LayerNorm fused via LDS transpose.
// ---------------------------------------------------------------------------

#define TOPK 30
#define HID 128

typedef __attribute__((ext_vector_type(2))) float v2f;
typedef __attribute__((ext_vector_type(8))) float v8f;

// ----------------------------- helpers -------------------------------------
__device__ __forceinline__ void cross3(const float a[3], const float b[3], float o[3]) {
    o[0] = a[1] * b[2] - a[2] * b[1];
    o[1] = a[2] * b[0] - a[0] * b[2];
    o[2] = a[0] * b[1] - a[1] * b[0];
}
__device__ __forceinline__ float dot3(const float a[3], const float b[3]) {
    return a[0] * b[0] + a[1] * b[1] + a[2] * b[2];
}
__device__ __forceinline__ void l2n3(float v[3]) {
    float n = sqrtf(dot3(v, v));
    n = fmaxf(n, 1e-12f);
    float inv = 1.f / n;
    v[0] *= inv; v[1] *= inv; v[2] *= inv;
}

// ---------------------------------------------------------------------------
// Kernel 1: per-batch sums  S[b][c]   = sum_n 1/|X[n,0,c]-X[n,1,c]|
//                           S[b][3+c] = sum_n 1/|X[n,2,c]-X[n,1,c]|
// (the axis=1 reduction inside _norm_pneg1_dim1)
// ---------------------------------------------------------------------------
__global__ void pf_vecsum_kernel(const float* __restrict__ X, float* __restrict__ S, int N) {
    const int b = blockIdx.x;
    const int t = threadIdx.x;
    float s0[3] = {0.f, 0.f, 0.f}, s1[3] = {0.f, 0.f, 0.f};
    const float* Xb = X + (size_t)b * N * 12;
    for (int n = t; n < N; n += 256) {
        const float* r = Xb + (size_t)n * 12;
#pragma unroll
        for (int c = 0; c < 3; ++c) {
            s0[c] += 1.f / fabsf(r[0 + c] - r[3 + c]);
            s1[c] += 1.f / fabsf(r[6 + c] - r[3 + c]);
        }
    }
    __shared__ float red[6][256];
#pragma unroll
    for (int c = 0; c < 3; ++c) { red[c][t] = s0[c]; red[3 + c][t] = s1[c]; }
    __syncthreads();
    for (int s = 128; s > 0; s >>= 1) {
        if (t < s) {
#pragma unroll
            for (int c = 0; c < 6; ++c) red[c][t] += red[c][t + s];
        }
        __syncthreads();
    }
    if (t < 6) S[b * 8 + t] = red[t][0];
}

// ---------------------------------------------------------------------------
// Kernel 2: per-residue frames O (B,N,3,3) and node features (dihedrals)
// Vf rows are padded 6 -> 8 for the WMMA K dimension.
// ---------------------------------------------------------------------------
__global__ void pf_node_kernel(const float* __restrict__ X, const float* __restrict__ S,
                               float* __restrict__ O, float* __restrict__ Vf, int N) {
    const int idx = blockIdx.x * 256 + threadIdx.x;  // b*N + n
    if (idx >= 2 * N) return;
    const int b = idx / N;
    const int n = idx % N;
    const float* r = X + (size_t)idx * 12;
    const float* Sb = S + b * 8;

    float v0[3], v1[3], vc[3];
#pragma unroll
    for (int c = 0; c < 3; ++c) {
        float nm0 = 1.f / Sb[c];
        float nm1 = 1.f / Sb[3 + c];
        v0[c] = (r[0 + c] - r[3 + c]) / fmaxf(nm0, 1e-6f);
        v1[c] = (r[6 + c] - r[3 + c]) / fmaxf(nm1, 1e-6f);
    }
    cross3(v0, v1, vc);
    float* Or = O + (size_t)idx * 9;
#pragma unroll
    for (int c = 0; c < 3; ++c) { Or[c] = v0[c]; Or[3 + c] = v1[c]; Or[6 + c] = vc[c]; }

    // dihedrals: backbone chain Xb[a] = X[b, a/3, a%3, :], a in [0, 3N)
    float dih[3];
    const float* Xbase = X + (size_t)b * N * 12;
#pragma unroll
    for (int p = 0; p < 3; ++p) {
        const int g = 3 * n + p;  // index into padded dihedral array
        if (g == 0 || g >= 3 * N - 2) { dih[p] = 0.f; continue; }
        const int t0 = g - 1;
        float U[3][3];
#pragma unroll
        for (int q = 0; q < 3; ++q) {
            const int aq = t0 + q;
            const float* A0 = Xbase + ((size_t)(aq / 3) * 4 + (aq % 3)) * 3;
            const float* A1 = Xbase + ((size_t)((aq + 1) / 3) * 4 + ((aq + 1) % 3)) * 3;
            U[q][0] = A1[0] - A0[0]; U[q][1] = A1[1] - A0[1]; U[q][2] = A1[2] - A0[2];
            l2n3(U[q]);
        }
        float n2[3], n1[3];
        cross3(U[0], U[1], n2); l2n3(n2);
        cross3(U[1], U[2], n1); l2n3(n1);
        float cd = dot3(n2, n1);
        cd = fminf(fmaxf(cd, -1.f + 1e-7f), 1.f - 1e-7f);
        float sg = dot3(U[0], n1);
        float sgn = (sg > 0.f) ? 1.f : ((sg < 0.f) ? -1.f : 0.f);
        dih[p] = sgn * acosf(cd);
    }
    float* vf = Vf + (size_t)idx * 8;
    vf[0] = cosf(dih[0]); vf[1] = cosf(dih[1]); vf[2] = cosf(dih[2]);
    vf[3] = sinf(dih[0]); vf[4] = sinf(dih[1]); vf[5] = sinf(dih[2]);
    vf[6] = 0.f; vf[7] = 0.f;
}

// ---------------------------------------------------------------------------
// Kernel 3: distances + top-30 per row. One block (256 thr) per (b,i) row.
// Tie-break: lowest index (matches jax.lax.top_k).
// ---------------------------------------------------------------------------
__global__ void pf_topk_kernel(const float* __restrict__ X, const float* __restrict__ mask,
                               float* __restrict__ Dnb, int* __restrict__ Eidx, int N) {
    const int bi = blockIdx.x;  // b*N + i
    const int b = bi / N;
    const int i = bi % N;
    const int t = threadIdx.x;
    __shared__ float dist[4096];
    __shared__ float redv[256];
    __shared__ int   redi[256];

    const float* Xb = X + (size_t)b * N * 12;
    const float xi0 = Xb[(size_t)i * 12 + 3], xi1 = Xb[(size_t)i * 12 + 4], xi2 = Xb[(size_t)i * 12 + 5];
    const float mi = mask[(size_t)b * N + i];

    float lmax = -1e30f;
    for (int j = t; j < N; j += 256) {
        float dx = Xb[(size_t)j * 12 + 3] - xi0;
        float dy = Xb[(size_t)j * 12 + 4] - xi1;
        float dz = Xb[(size_t)j * 12 + 5] - xi2;
        float d2 = fmaxf(dx * dx + dy * dy + dz * dz, 0.f);
        float m = mi * mask[(size_t)b * N + j];
        float d = m * sqrtf(d2 + 1e-6f);
        dist[j] = d;
        lmax = fmaxf(lmax, d);
    }
    redv[t] = lmax;
    __syncthreads();
    for (int s = 128; s > 0; s >>= 1) {
        if (t < s) redv[t] = fmaxf(redv[t], redv[t + s]);
        __syncthreads();
    }
    const float Dmax = redv[0];
    __syncthreads();
    for (int j = t; j < N; j += 256) {
        float m = mi * mask[(size_t)b * N + j];
        dist[j] += (1.f - m) * Dmax;
    }
    __syncthreads();

    for (int k = 0; k < TOPK; ++k) {
        float bv = 3.4e38f;
        int bj = N;
        for (int j = t; j < N; j += 256) {
            float v = dist[j];
            if (v < bv || (v == bv && j < bj)) { bv = v; bj = j; }
        }
        redv[t] = bv; redi[t] = bj;
        __syncthreads();
        for (int s = 128; s > 0; s >>= 1) {
            if (t < s) {
                float v2 = redv[t + s]; int j2 = redi[t + s];
                if (v2 < redv[t] || (v2 == redv[t] && j2 < redi[t])) { redv[t] = v2; redi[t] = j2; }
            }
            __syncthreads();
        }
        if (t == 0) {
            Eidx[(size_t)bi * TOPK + k] = redi[0];
            Dnb[(size_t)bi * TOPK + k] = redv[0];
            dist[redi[0]] = 3.4e38f;
        }
        __syncthreads();
    }
}

// ---------------------------------------------------------------------------
// Kernel 4: per-edge 39-dim feature assembly -> Ef (padded to 40 cols)
// order: pos-embed(16) | RBF(16) | dU(3) | quat(4) | pad(1)
// ---------------------------------------------------------------------------
__global__ void pf_edgefeat_kernel(const float* __restrict__ X, const int* __restrict__ srr,
                                   const float* __restrict__ O, const float* __restrict__ Dnb,
                                   const int* __restrict__ Eidx, float* __restrict__ Ef, int N) {
    const long long e = (long long)blockIdx.x * 256 + threadIdx.x;  // ((b*N+i)*30+k)
    const long long M = 2LL * N * TOPK;
    if (e >= M) return;
    const int bi = (int)(e / TOPK);  // b*N + i
    const int b = bi / N;
    const int j = Eidx[e];
    const float d = Dnb[e];

    float f[40];
    // positional embedding of rel = srr[b, j]
    const float rel = (float)srr[(size_t)b * N + j];
    const float negln = -9.210340371976184f;  // -ln(10000)
#pragma unroll
    for (int q = 0; q < 8; ++q) {
        float div = expf(negln * (2.f * q) / 16.f);
        float ang = rel * div;
        f[2 * q]     = sinf(ang);
        f[2 * q + 1] = cosf(ang);
    }
    // RBF
#pragma unroll
    for (int m = 0; m < 16; ++m) {
        float mu = (20.f / 15.f) * m;
        float z = (d - mu) / 1.25f;
        f[16 + m] = expf(-z * z);
    }
    // orientation features
    const float* Oi = O + (size_t)bi * 9;
    const float* Oj = O + ((size_t)b * N + j) * 9;
    const float* Xi = X + (size_t)bi * 12 + 3;
    const float* Xj = X + ((size_t)b * N + j) * 12 + 3;
    float dX[3] = {Xj[0] - Xi[0], Xj[1] - Xi[1], Xj[2] - Xi[2]};
    float dU[3];
#pragma unroll
    for (int c = 0; c < 3; ++c)
        dU[c] = Oi[c * 3 + 0] * dX[0] + Oi[c * 3 + 1] * dX[1] + Oi[c * 3 + 2] * dX[2];
    l2n3(dU);
    f[32] = dU[0]; f[33] = dU[1]; f[34] = dU[2];

    // R = Oi^T * Oj
    float R[3][3];
#pragma unroll
    for (int r0 = 0; r0 < 3; ++r0)
#pragma unroll
        for (int c0 = 0; c0 < 3; ++c0)
            R[r0][c0] = Oi[0 * 3 + r0] * Oj[0 * 3 + c0] + Oi[1 * 3 + r0] * Oj[1 * 3 + c0] +
                        Oi[2 * 3 + r0] * Oj[2 * 3 + c0];
    // quaternion (principal eigenvector of the Davenport K matrix, branch method)
    float q0, q1, q2, q3;
    const float tr = R[0][0] + R[1][1] + R[2][2];
    if (tr > 0.f) {
        float s = sqrtf(fmaxf(tr + 1.f, 1e-8f)) * 2.f;
        q0 = 0.25f * s;
        q1 = (R[2][1] - R[1][2]) / s;
        q2 = (R[0][2] - R[2][0]) / s;
        q3 = (R[1][0] - R[0][1]) / s;
    } else if (R[0][0] > R[1][1] && R[0][0] > R[2][2]) {
        float s = sqrtf(fmaxf(1.f + R[0][0] - R[1][1] - R[2][2], 1e-8f)) * 2.f;
        q0 = (R[2][1] - R[1][2]) / s;
        q1 = 0.25f * s;
        q2 = (R[0][1] + R[1][0]) / s;
        q3 = (R[0][2] + R[2][0]) / s;
    } else if (R[1][1] > R[2][2]) {
        float s = sqrtf(fmaxf(1.f + R[1][1] - R[0][0] - R[2][2], 1e-8f)) * 2.f;
        q0 = (R[0][2] - R[2][0]) / s;
        q1 = (R[0][1] + R[1][0]) / s;
        q2 = 0.25f * s;
        q3 = (R[1][2] + R[2][1]) / s;
    } else {
        float s = sqrtf(fmaxf(1.f + R[2][2] - R[0][0] - R[1][1], 1e-8f)) * 2.f;
        q0 = (R[1][0] - R[0][1]) / s;
        q1 = (R[0][2] + R[2][0]) / s;
        q2 = (R[1][2] + R[2][1]) / s;
        q3 = 0.25f * s;
    }
    float qn = sqrtf(fmaxf(q0 * q0 + q1 * q1 + q2 * q2 + q3 * q3, 1e-12f));
    f[35] = q0 / qn; f[36] = q1 / qn; f[37] = q2 / qn; f[38] = q3 / qn;
    f[39] = 0.f;  // K-pad

    float* out = Ef + (size_t)e * 40;
#pragma unroll
    for (int c = 0; c < 40; ++c) out[c] = f[c];
}

// ---------------------------------------------------------------------------
// Kernel 5: fused GEMM (V_WMMA_F32_16X16X4_F32) + bias + LayerNorm(ddof=1).
// A: M x KP row-major (zero-padded), W: KREAL x 128 row-major.
// One wave per 16-row tile; 4 waves / block; M must be multiple of 64.
// A fragment (16x4 f32): lane m = lane&15; VGPRs hold K pairs split by lane
// half -> each lane loads float2 at A[row][kbase + 2*(lane>>4)].
// B fragment (4x16 f32): mirrored layout over N = lane&15.
// ---------------------------------------------------------------------------
template <int KP, int KSTEPS, int KREAL>
__global__ void pf_gemm_ln_kernel(const float* __restrict__ A, const float* __restrict__ W,
                                  const float* __restrict__ bvec, const float* __restrict__ gain,
                                  const float* __restrict__ bias, float* __restrict__ out) {
    __shared__ float lds[4][16 * HID];
    const int wave = threadIdx.x >> 5;
    const int lane = threadIdx.x & 31;
    const int mt = blockIdx.x * 4 + wave;
    const int row0 = mt * 16;
    const int m = lane & 15;
    const int h = lane >> 4;

    __builtin_prefetch(W, 0, 0);  // global_prefetch_b8 hint for the shared weights

    v2f a[KSTEPS];
#pragma unroll
    for (int s = 0; s < KSTEPS; ++s) {
        const float* p = A + (size_t)(row0 + m) * KP + s * 4 + 2 * h;
        a[s].x = p[0];
        a[s].y = p[1];
    }

    float* myl = &lds[wave][0];
#pragma unroll
    for (int ct = 0; ct < 8; ++ct) {
        v8f acc = {};
#pragma unroll
        for (int s = 0; s < KSTEPS; ++s) {
            const int k0 = s * 4 + 2 * h;
            v2f bb;
            bb.x = (k0 < KREAL) ? W[(size_t)k0 * HID + ct * 16 + m] : 0.f;
            bb.y = (k0 + 1 < KREAL) ? W[(size_t)(k0 + 1) * HID + ct * 16 + m] : 0.f;
            acc = __builtin_amdgcn_wmma_f32_16x16x4_f32(
                /*neg_a=*/false, a[s], /*neg_b=*/false, bb,
                /*c_mod=*/(short)0, acc, /*reuse_a=*/false, /*reuse_b=*/false);
        }
        // D layout: VGPR v holds rows v (lanes 0-15) and v+8 (lanes 16-31), N = lane&15
#pragma unroll
        for (int v = 0; v < 8; ++v) myl[(v + 8 * h) * HID + ct * 16 + m] = acc[v];
    }
    __syncthreads();

    // fused bias + LayerNorm: lanes 0..15 each own one row of the tile
    if (lane < 16) {
        const float* r = myl + lane * HID;
        float mu = 0.f;
        for (int c = 0; c < HID; ++c) mu += r[c] + bvec[c];
        mu *= (1.f / HID);
        float var = 0.f;
        for (int c = 0; c < HID; ++c) {
            float dd = r[c] + bvec[c] - mu;
            var += dd * dd;
        }
        float sigma = sqrtf(var * (1.f / (HID - 1)));
        float inv = 1.f / (sigma + 1e-6f);
        float* o = out + (size_t)(row0 + lane) * HID;
        for (int c = 0; c < HID; ++c)
            o[c] = gain[c] * ((r[c] + bvec[c]) - mu) * inv + bias[c];
    }
}

// ---------------------------------------------------------------------------
// launch
// ---------------------------------------------------------------------------
extern "C" void kernel_launch(void* const* d_in, const int* in_sizes, int n_in,
                              void* d_out, int out_size, void* d_ws, size_t ws_size,
                              hipStream_t stream) {
    const float* X    = (const float*)d_in[0];
    const float* mask = (const float*)d_in[1];
    const int*   srr  = (const int*)d_in[2];
    const float* Wn   = (const float*)d_in[3];
    const float* bn   = (const float*)d_in[4];
    const float* We   = (const float*)d_in[5];
    const float* be   = (const float*)d_in[6];
    const float* gn   = (const float*)d_in[7];
    const float* bsn  = (const float*)d_in[8];
    const float* ge   = (const float*)d_in[9];
    const float* bse  = (const float*)d_in[10];

    const int B = 2;
    const int N = in_sizes[1] / B;          // 4096
    const long long BN = (long long)B * N;  // 8192
    const long long M_edge = BN * TOPK;     // 245760

    // workspace layout (floats)
    float* ws   = (float*)d_ws;
    float* wsS  = ws;                         // 16
    float* wsO  = wsS + 16;                   // BN*9
    float* wsD  = wsO + BN * 9;               // BN*TOPK
    float* wsVf = wsD + M_edge;               // BN*8
    float* wsEf = wsVf + BN * 8;              // M_edge*40

    // d_out layout: V | E | E_idx (int bits)
    float* outV = (float*)d_out;
    float* outE = outV + BN * HID;
    int*   outI = (int*)(outE + M_edge * HID);

    pf_vecsum_kernel<<<B, 256, 0, stream>>>(X, wsS, N);
    pf_node_kernel<<<(int)((BN + 255) / 256), 256, 0, stream>>>(X, wsS, wsO, wsVf, N);
    pf_topk_kernel<<<(int)BN, 256, 0, stream>>>(X, mask, wsD, outI, N);
    pf_edgefeat_kernel<<<(int)((M_edge + 255) / 256), 256, 0, stream>>>(X, srr, wsO, wsD, outI, wsEf, N);
    // edge GEMM: M=245760 -> 15360 tiles -> 3840 blocks of 4 waves
    pf_gemm_ln_kernel<40, 10, 39><<<(int)(M_edge / 64), 128, 0, stream>>>(wsEf, We, be, ge, bse, outE);
    // node GEMM: M=8192 -> 512 tiles -> 128 blocks
    pf_gemm_ln_kernel<8, 2, 6><<<(int)(BN / 64), 128, 0, stream>>>(wsVf, Wn, bn, gn, bsn, outV);
}